// sysRolloutPolicy_9723805958638
// MI455X (gfx1250) — compile-verified
//
#include <hip/hip_runtime.h>
#include <hip/hip_bf16.h>
#include <stdint.h>

// Problem geometry (from the reference)
#define N_AGENTS 8192
#define THREADS  256
#define CHUNK    512                    // goals staged per workgroup via TDM
#define NCHUNKS  (N_AGENTS / CHUNK)     // 16

typedef __attribute__((ext_vector_type(4))) unsigned int v4u;
typedef __attribute__((ext_vector_type(8))) int          v8i;
typedef __attribute__((ext_vector_type(4))) int          v4i;

// ---------------------------------------------------------------------------
// Kernel 0: initialize the packed (dist_bits, idx) argmin workspace to +inf.
// d_ws is poisoned once and NOT re-poisoned between replays, so we must
// rewrite it every call.
// ---------------------------------------------------------------------------
__global__ __launch_bounds__(THREADS)
void nn_init_kernel(unsigned long long* __restrict__ best) {
    best[blockIdx.x * THREADS + threadIdx.x] = ~0ull;
}

// ---------------------------------------------------------------------------
// Kernel 1: partial L1-argmin. Grid = (agent_tiles=32, goal_chunks=16).
// Each workgroup stages its 512-goal chunk into LDS with the Tensor Data
// Mover (strided 12B rows -> packed float2 tile), then each thread scans the
// chunk for its agent and merges via a u64 atomic-min whose key is
// (float_bits(dist) << 32) | goal_index  — this preserves jnp.argmin's
// first-index tiebreak because dist >= 0 (monotone bit pattern) and smaller
// global index wins on equal distance.
// ---------------------------------------------------------------------------
__global__ __launch_bounds__(THREADS)
void nn_argmin_kernel(const float* __restrict__ x,
                      unsigned long long* __restrict__ best) {
    __shared__ float2 sg[CHUNK];        // 4 KB packed goal tile

    const int agent = blockIdx.x * THREADS + threadIdx.x;
    const int chunk = blockIdx.y;

    // --- TDM stage: one wave issues the tensor DMA (EXEC is ignored by TDM,
    // but we gate so only wave 0 launches it; redundant DMAs would be waste).
    if (threadIdx.x < 32) {
        // Tile start = first element of goal row (chunk*CHUNK) in the
        // (16384 x 3) row-major array; goals begin at row N_AGENTS.
        unsigned long long gaddr =
            (unsigned long long)(uintptr_t)(x + 3ull * (N_AGENTS + (unsigned long long)chunk * CHUNK));
        // Generic LDS pointer: low 32 bits are the LDS byte offset.
        unsigned lds_addr = (unsigned)(uintptr_t)(&sg[0]);

        // D# group 0 (128b): count=1 | lds_addr | global_addr[56:0] | type=2
        v4u g0 = { 1u,
                   lds_addr,
                   (unsigned)(gaddr & 0xffffffffu),
                   (unsigned)((gaddr >> 32) & 0x01ffffffu) | (2u << 30) };

        // D# group 1 (256b):
        //  w0: workgroup_mask=0, data_size=2 (4B), no barrier/iterate/pad
        //  tensor_dim0=3 (row length), tensor_dim1=CHUNK rows
        //  tile_dim0=2 (take x,y), tile_dim1=CHUNK, tile_dim2=0
        //  tensor_dim0_stride=3, tensor_dim1_stride=0 (2-D tile)
        v8i g1 = { (int)(2u << 16),                 // data_size = 4 bytes
                   (int)(3u << 16),                 // tensor_dim0 = 3 (lo16 @ bit48)
                   (int)((unsigned)CHUNK << 16),    // tensor_dim1 lo16 @ bit80
                   (int)(2u << 16),                 // tile_dim0 = 2 @ bit112
                   (int)CHUNK,                      // tile_dim1 @ bit128
                   3,                               // tensor_dim0_stride @ bit160
                   0, 0 };

        v4i g2 = { 0, 0, 0, 0 };                    // dims 2/3 unused (zero)
        v4i g3 = { 0, 0, 0, 0 };
        v8i g4 = { 0, 0, 0, 0, 0, 0, 0, 0 };        // trailing group (unused, zero)

        // clang-23 / therock-10.0 form: 6 args
        __builtin_amdgcn_tensor_load_to_lds(g0, g1, g2, g3, g4, /*cpol=*/0);
        __builtin_amdgcn_s_wait_tensorcnt(0);
    }
    __syncthreads();

    const float ax = x[3 * agent];
    const float ay = x[3 * agent + 1];

    float bestd = __FLT_MAX__;
    int   bestj = 0;
#pragma unroll 8
    for (int j = 0; j < CHUNK; ++j) {
        float2 g = sg[j];                            // LDS broadcast read
        float  d = fabsf(ax - g.x) + fabsf(ay - g.y);
        if (d < bestd) { bestd = d; bestj = j; }     // strict '<' => first min
    }

    unsigned long long key =
        ((unsigned long long)__float_as_uint(bestd) << 32)
        | (unsigned)(chunk * CHUNK + bestj);
    atomicMin(&best[agent], key);                    // global_atomic_min_u64
}

// ---------------------------------------------------------------------------
// Kernel 2: decode argmin index, replicate the reference's prob/cdf/argmax
// chain exactly in f32 (small-integer arithmetic is exact, and the >=
// threshold chain reproduces argmax-first-true semantics, incl. u == 0).
// ---------------------------------------------------------------------------
__global__ __launch_bounds__(THREADS)
void nn_action_kernel(const float* __restrict__ x,
                      const float* __restrict__ u,
                      const unsigned long long* __restrict__ best,
                      int* __restrict__ out) {
    const int i = blockIdx.x * THREADS + threadIdx.x;
    const unsigned j = (unsigned)(best[i] & 0xffffffffull);

    const float ax = x[3 * i];
    const float ay = x[3 * i + 1];
    const float gx = x[3u * (N_AGENTS + j)];
    const float gy = x[3u * (N_AGENTS + j) + 1];

    const float dx = gx - ax;
    const float dy = gy - ay;

    const float p0 = (dx > 0.0f) ? 1.0f : 0.0f;
    const float p1 = (dx < 0.0f) ? 1.0f : 0.0f;
    const float p2 = (dy > 0.0f) ? 1.0f : 0.0f;
    const float p3 = (dy < 0.0f) ? 1.0f : 0.0f;
    const float p4 = (dx == 0.0f && dy == 0.0f) ? 1.0f : 0.0f;

    const float c0 = p0;
    const float c1 = c0 + p1;
    const float c2 = c1 + p2;
    const float c3 = c2 + p3;
    const float c4 = c3 + p4;

    const float t = u[i] * c4;
    int a;
    if      (c0 >= t) a = 0;
    else if (c1 >= t) a = 1;
    else if (c2 >= t) a = 2;
    else if (c3 >= t) a = 3;
    else              a = 4;

    out[i] = a;
}

// ---------------------------------------------------------------------------
extern "C" void kernel_launch(void* const* d_in, const int* in_sizes, int n_in,
                              void* d_out, int out_size, void* d_ws, size_t ws_size,
                              hipStream_t stream) {
    (void)in_sizes; (void)n_in; (void)out_size; (void)ws_size;

    const float* x = (const float*)d_in[0];   // (16384, 3) f32
    const float* u = (const float*)d_in[1];   // (8192,)   f32
    int* out = (int*)d_out;                   // 8192 int32 actions
    unsigned long long* best = (unsigned long long*)d_ws;  // 64 KB scratch

    nn_init_kernel<<<N_AGENTS / THREADS, THREADS, 0, stream>>>(best);
    nn_argmin_kernel<<<dim3(N_AGENTS / THREADS, NCHUNKS), THREADS, 0, stream>>>(x, best);
    nn_action_kernel<<<N_AGENTS / THREADS, THREADS, 0, stream>>>(x, u, best, out);
}